// EfficientDet_76819785056363
// MI455X (gfx1250) — compile-verified
//
#include <hip/hip_runtime.h>
#include <hip/hip_bf16.h>
#include <stdint.h>

// ---------------- problem constants (match reference) ----------------
#define B_  4
#define A_  49104
#define C_  90
#define PRE_NMS_ 1000
#define POST_NMS_ 100
#define IOU_THR_ 0.5f
#define SCORE_THR_ 0.05f
#define IMG_ 512.0f

// ---------------- helpers ----------------
__device__ __forceinline__ unsigned f2k(float f) {
    unsigned u = __float_as_uint(f);
    return (u & 0x80000000u) ? ~u : (u | 0x80000000u);
}
__device__ __forceinline__ float k2f(unsigned k) {
    unsigned u = (k & 0x80000000u) ? (k & 0x7FFFFFFFu) : ~k;
    return __uint_as_float(u);
}

// CDNA5 async global->LDS (ASYNCcnt path). Builtin takes pointers to 16-byte
// int vectors in AS(1)/AS(3). Falls back to a plain copy if unavailable.
typedef int v4i_async __attribute__((__vector_size__(4 * sizeof(int))));

__device__ __forceinline__ void async_copy_b128(const void* gsrc, void* lds_dst) {
#if __has_builtin(__builtin_amdgcn_global_load_async_to_lds_b128)
    __builtin_amdgcn_global_load_async_to_lds_b128(
        (__attribute__((address_space(1))) v4i_async*)(gsrc),
        (__attribute__((address_space(3))) v4i_async*)(lds_dst), 0, 0);
#else
    *(float4*)lds_dst = *(const float4*)gsrc;
#endif
}
__device__ __forceinline__ void async_wait_all() {
#if __has_builtin(__builtin_amdgcn_global_load_async_to_lds_b128)
    asm volatile("s_wait_asynccnt 0" ::: "memory");
#endif
}

// ---------------- kernel 1: anchors ----------------
__global__ void anchors_kernel(float4* __restrict__ anchors) {
    int a = blockIdx.x * blockDim.x + threadIdx.x;
    if (a >= A_) return;
    int lvl, off;
    if      (a < 36864) { lvl = 0; off = 0; }
    else if (a < 46080) { lvl = 1; off = 36864; }
    else if (a < 48384) { lvl = 2; off = 46080; }
    else if (a < 48960) { lvl = 3; off = 48384; }
    else                { lvl = 4; off = 48960; }
    int r = a - off;
    int feat = 64 >> lvl;
    float stride = (float)(8 << lvl);
    int k = r % 9;
    int cell = r / 9;
    int x = cell % feat, y = cell / feat;
    int s = k / 3, ri = k % 3;
    const float scales[3] = {4.0f, 5.0396842f, 6.3496042f};   // 4*2^(s/3)
    const float sqrtr[3]  = {0.70710678f, 1.0f, 1.41421356f}; // sqrt(ratio)
    float ws = stride * scales[s] * sqrtr[ri];
    float hs = stride * scales[s] / sqrtr[ri];
    float cy = (y + 0.5f) * stride, cx = (x + 0.5f) * stride;
    anchors[a] = make_float4(cy - 0.5f * hs, cx - 0.5f * ws,
                             cy + 0.5f * hs, cx + 0.5f * ws);
}

// ---------------- kernel 2: decode + normalize + clip ----------------
__global__ void decode_kernel(const float4* __restrict__ deltas,
                              const float4* __restrict__ anchors,
                              float4* __restrict__ boxes) {
    int g = blockIdx.x * blockDim.x + threadIdx.x;  // over B*A
    if (g >= B_ * A_) return;
    int a = g % A_;
    float4 an = anchors[a];
    float4 d  = deltas[g];
    float ah = an.z - an.x, aw = an.w - an.y;
    float acy = 0.5f * (an.x + an.z), acx = 0.5f * (an.y + an.w);
    float dy = d.x * 0.1f, dx = d.y * 0.1f, dh = d.z * 0.2f, dw = d.w * 0.2f;
    float cy = dy * ah + acy, cx = dx * aw + acx;
    float h = expf(dh) * ah, w = expf(dw) * aw;
    const float inv = 1.0f / IMG_;
    float4 o;
    o.x = fminf(fmaxf((cy - 0.5f * h) * inv, 0.f), 1.f);
    o.y = fminf(fmaxf((cx - 0.5f * w) * inv, 0.f), 1.f);
    o.z = fminf(fmaxf((cy + 0.5f * h) * inv, 0.f), 1.f);
    o.w = fminf(fmaxf((cx + 0.5f * w) * inv, 0.f), 1.f);
    boxes[g] = o;
}

// ---------------- kernel 3: transpose logits -> sortable keys [B,C,A] ----------
__global__ void transpose_key_kernel(const float* __restrict__ labels,
                                     unsigned* __restrict__ keysT) {
    __shared__ unsigned tile[32][33];
    int b  = blockIdx.z;
    int a0 = blockIdx.x * 32;
    int c0 = blockIdx.y * 32;
    int tx = threadIdx.x;   // 0..31 (along c on read, along a on write)
    int ty = threadIdx.y;   // 0..7
    const float* src = labels + (size_t)b * A_ * C_;
    unsigned* dst = keysT + (size_t)b * C_ * A_;
    #pragma unroll
    for (int i = 0; i < 32; i += 8) {
        int a = a0 + ty + i, c = c0 + tx;
        if (a < A_ && c < C_) tile[ty + i][tx] = f2k(src[(size_t)a * C_ + c]);
    }
    __syncthreads();
    #pragma unroll
    for (int i = 0; i < 32; i += 8) {
        int c = c0 + ty + i, a = a0 + tx;
        if (c < C_ && a < A_) dst[(size_t)c * A_ + a] = tile[tx][ty + i];
    }
}

// ---------------- kernel 4: per-(b,c) radix-select top-1000 + bitonic sort ----
#define TOPK_THREADS 256
__global__ void topk_kernel(const unsigned* __restrict__ keysT,
                            const float4* __restrict__ boxes,
                            float* __restrict__ topv,
                            float4* __restrict__ topboxes) {
    const int bc = blockIdx.x;           // 0..B*C-1
    const int b  = bc / C_;
    const unsigned* keys = keysT + (size_t)bc * A_;
    __shared__ unsigned hist[256];
    __shared__ unsigned sh_prefix;
    __shared__ int sh_remK, sh_cg, sh_ce;
    __shared__ unsigned long long arr[1024];
    const int tid = threadIdx.x;
    if (tid == 0) { sh_prefix = 0u; sh_remK = PRE_NMS_; sh_cg = 0; sh_ce = 0; }
    // 4-pass MSB-first radix select (keys are order-preserving uint of logits)
    for (int pass = 0; pass < 4; ++pass) {
        hist[tid] = 0;
        __syncthreads();
        const int shift = 24 - 8 * pass;
        const unsigned mask = (pass == 0) ? 0u : (0xFFFFFFFFu << (32 - 8 * pass));
        const unsigned prefix = sh_prefix;
        for (int i = tid; i < A_; i += TOPK_THREADS) {
            unsigned k = keys[i];
            if ((k & mask) == prefix) atomicAdd(&hist[(k >> shift) & 255], 1u);
        }
        __syncthreads();
        if (tid == 0) {
            unsigned cum = 0; int remK = sh_remK; int sel = 0;
            for (int bb = 255; bb >= 0; --bb) {
                unsigned h = hist[bb];
                if (cum + h >= (unsigned)remK) { sel = bb; break; }
                cum += h;
            }
            sh_remK = remK - (int)cum;
            sh_prefix = prefix | ((unsigned)sel << shift);
        }
        __syncthreads();
    }
    const unsigned T = sh_prefix;
    const int takeEq = sh_remK;
    const int G = PRE_NMS_ - takeEq;     // exact count of keys strictly > T
    // compact (key,idx) into LDS; greater fill [0,G), equals fill [G, 1000)
    for (int i = tid; i < A_; i += TOPK_THREADS) {
        unsigned k = keys[i];
        if (k > T) {
            int p = atomicAdd(&sh_cg, 1);
            if (p < PRE_NMS_)
                arr[p] = ((unsigned long long)k << 32) | (unsigned)(~(unsigned)i);
        } else if (k == T) {
            int q = atomicAdd(&sh_ce, 1);
            if (q < takeEq)
                arr[G + q] = ((unsigned long long)k << 32) | (unsigned)(~(unsigned)i);
        }
    }
    for (int i = PRE_NMS_ + tid; i < 1024; i += TOPK_THREADS) arr[i] = 0ull;
    __syncthreads();
    // bitonic sort 1024, descending (key desc, index asc via ~idx in low bits)
    for (unsigned kk = 2; kk <= 1024; kk <<= 1) {
        for (unsigned j = kk >> 1; j > 0; j >>= 1) {
            for (unsigned i = tid; i < 1024; i += TOPK_THREADS) {
                unsigned ixj = i ^ j;
                if (ixj > i) {
                    unsigned long long x = arr[i], y = arr[ixj];
                    bool up = ((i & kk) == 0);
                    if (up ? (x < y) : (x > y)) { arr[i] = y; arr[ixj] = x; }
                }
            }
            __syncthreads();
        }
    }
    // emit sigmoid(score) + gathered boxes
    for (int t = tid; t < PRE_NMS_; t += TOPK_THREADS) {
        unsigned long long v = arr[t];
        unsigned k = (unsigned)(v >> 32);
        unsigned idx = ~((unsigned)v);
        float logit = k2f(k);
        float prob = 1.0f / (1.0f + expf(-logit));
        topv[(size_t)bc * PRE_NMS_ + t] = prob;
        topboxes[(size_t)bc * PRE_NMS_ + t] = boxes[(size_t)b * A_ + idx];
    }
}

// ---------------- kernel 5: bitmask NMS (320KB-LDS formulation) ----------------
// Phase 1 (parallel, barrier-free): 1000x1000 suppression bit-matrix in LDS.
// Phase 2 (wave32 lockstep, no barriers): greedy scan, keep-word per lane.
#define NMS_THREADS 256
__global__ void __launch_bounds__(NMS_THREADS)
nms_kernel(const float4* __restrict__ topboxes,
           const float* __restrict__ topv,
           float* __restrict__ cand_s,
           int* __restrict__ cand_i) {
    const int bc = blockIdx.x;
    const int c  = bc % C_;
    __shared__ alignas(16) float4 lbox[PRE_NMS_];       //  16 KB
    __shared__ alignas(16) float  lsc[PRE_NMS_];        //   4 KB
    __shared__ unsigned supmat[PRE_NMS_][33];           // 132 KB (row pad: no bank conflicts)
    __shared__ unsigned keepw[32];
    const int tid = threadIdx.x;
    const float4* gsrc = topboxes + (size_t)bc * PRE_NMS_;
    const float4* ssrc = (const float4*)(topv + (size_t)bc * PRE_NMS_);
    // async global->LDS staging (GLOBAL_LOAD_ASYNC_TO_LDS_B128, ASYNCcnt)
    for (int i = tid; i < PRE_NMS_; i += NMS_THREADS)
        async_copy_b128(gsrc + i, &lbox[i]);
    for (int i = tid; i < PRE_NMS_ / 4; i += NMS_THREADS)
        async_copy_b128(ssrc + i, ((float4*)lsc) + i);
    async_wait_all();
    __syncthreads();
    // phase 1: row-parallel suppression matrix (division-free IoU test)
    for (int i = tid; i < PRE_NMS_; i += NMS_THREADS) {
        float4 bi = lbox[i];
        float a1 = (bi.z - bi.x) * (bi.w - bi.y);
        const int w0 = i >> 5;
        for (int w = 0; w < w0; ++w) supmat[i][w] = 0u;
        for (int w = w0; w < 32; ++w) {
            unsigned m = 0u;
            const int jbase = w << 5;
            for (int t = 0; t < 32; ++t) {
                int j = jbase + t;
                if (j > i && j < PRE_NMS_) {
                    float4 bj = lbox[j];
                    float yy1 = fmaxf(bi.x, bj.x), xx1 = fmaxf(bi.y, bj.y);
                    float yy2 = fminf(bi.z, bj.z), xx2 = fminf(bi.w, bj.w);
                    float inter = fmaxf(yy2 - yy1, 0.f) * fmaxf(xx2 - xx1, 0.f);
                    float a2 = (bj.z - bj.x) * (bj.w - bj.y);
                    // inter/(a1+a2-inter+eps) > thr  <=>  inter > thr*(denom), denom>0
                    if (inter > IOU_THR_ * (a1 + a2 - inter + 1e-8f)) m |= (1u << t);
                }
            }
            supmat[i][w] = m;
        }
    }
    __syncthreads();
    // phase 2: greedy scan in one wave32; lane L owns keep bits [32L, 32L+31]
    if (tid < 32) {
        unsigned kw = 0xFFFFFFFFu;
        for (int i = 0; i < PRE_NMS_; ++i) {
            unsigned wi = __shfl(kw, i >> 5, 32);    // broadcast owner's word
            if ((wi >> (i & 31)) & 1u)               // uniform branch
                kw &= ~supmat[i][tid];
        }
        keepw[tid] = kw;
    }
    __syncthreads();
    // per-class candidate list: first <=100 kept entries (already score-sorted)
    if (tid == 0) {
        float* cs = cand_s + (size_t)bc * POST_NMS_;
        int*   ci = cand_i + (size_t)bc * POST_NMS_;
        int cnt = 0;
        for (int j = 0; j < PRE_NMS_ && cnt < POST_NMS_; ++j) {
            if ((keepw[j >> 5] >> (j & 31)) & 1u) {
                float s = lsc[j];
                if (s > SCORE_THR_) { cs[cnt] = s; ci[cnt] = c * PRE_NMS_ + j; ++cnt; }
            }
        }
        for (; cnt < POST_NMS_; ++cnt) { cs[cnt] = -1.0f; ci[cnt] = 0; }
    }
}

// ---------------- kernel 6: per-batch top-100 + output assembly ----------------
#define FIN_THREADS 256
__global__ void final_topk_kernel(const float* __restrict__ cand_s,
                                  const int* __restrict__ cand_i,
                                  const float4* __restrict__ topboxes,
                                  float* __restrict__ out) {
    const int b = blockIdx.x;
    const int M = C_ * POST_NMS_;              // 9000
    __shared__ float s_sh[C_ * POST_NMS_];     // 36 KB
    __shared__ float red_s[FIN_THREADS];
    __shared__ int   red_p[FIN_THREADS];
    const int tid = threadIdx.x;
    const float* cs = cand_s + (size_t)b * M;
    for (int i = tid; i < M; i += FIN_THREADS) s_sh[i] = cs[i];
    __syncthreads();
    float* boxes_out   = out;                                  // [B][100][4]
    float* scores_out  = out + B_ * POST_NMS_ * 4;             // +1600
    float* classes_out = scores_out + B_ * POST_NMS_;          // +400
    int*   valid_out   = (int*)(classes_out + B_ * POST_NMS_); // +400
    int valid = 0;
    for (int r = 0; r < POST_NMS_; ++r) {
        float bs = -3.0f; int bp = 0;
        for (int i = tid; i < M; i += FIN_THREADS) {
            float s = s_sh[i];
            if (s > bs) { bs = s; bp = i; }
        }
        red_s[tid] = bs; red_p[tid] = bp;
        __syncthreads();
        for (int off = FIN_THREADS / 2; off > 0; off >>= 1) {
            if (tid < off) {
                float s2 = red_s[tid + off]; int p2 = red_p[tid + off];
                if (s2 > red_s[tid] || (s2 == red_s[tid] && p2 < red_p[tid])) {
                    red_s[tid] = s2; red_p[tid] = p2;
                }
            }
            __syncthreads();
        }
        bs = red_s[0]; bp = red_p[0];
        if (tid == 0) {
            float* bo = boxes_out + ((size_t)b * POST_NMS_ + r) * 4;
            if (bs > 0.f) {
                int fi  = cand_i[(size_t)b * M + bp];
                int cls = fi / PRE_NMS_, j = fi % PRE_NMS_;
                float4 bx = topboxes[((size_t)b * C_ + cls) * PRE_NMS_ + j];
                bo[0] = bx.x; bo[1] = bx.y; bo[2] = bx.z; bo[3] = bx.w;
                scores_out[b * POST_NMS_ + r]  = bs;
                classes_out[b * POST_NMS_ + r] = (float)cls;
                ++valid;
            } else {
                bo[0] = 0.f; bo[1] = 0.f; bo[2] = 0.f; bo[3] = 0.f;
                scores_out[b * POST_NMS_ + r]  = 0.f;
                classes_out[b * POST_NMS_ + r] = 0.f;
            }
            s_sh[bp] = -2.0f;   // mark taken
        }
        __syncthreads();
    }
    if (tid == 0) valid_out[b] = valid;
}

// ---------------- host-side launch ----------------
extern "C" void kernel_launch(void* const* d_in, const int* in_sizes, int n_in,
                              void* d_out, int out_size, void* d_ws, size_t ws_size,
                              hipStream_t stream) {
    (void)in_sizes; (void)n_in; (void)out_size; (void)ws_size;
    const float* deltas = (const float*)d_in[0];   // [B,A,4] f32
    const float* labels = (const float*)d_in[1];   // [B,A,C] f32

    // workspace layout (128-byte aligned, total ~82.2 MB)
    char* ws = (char*)d_ws;
    const size_t OFF_ANCH = 0;
    const size_t OFF_BOX  = OFF_ANCH + (size_t)A_ * 4 * 4;                 //   785,664
    const size_t OFF_KEYS = OFF_BOX  + (size_t)B_ * A_ * 4 * 4;            // 3,928,320
    const size_t OFF_TOPV = OFF_KEYS + (size_t)B_ * C_ * A_ * 4;           // 74,638,080
    const size_t OFF_TOPB = OFF_TOPV + (size_t)B_ * C_ * PRE_NMS_ * 4;     // 76,078,080
    const size_t OFF_CS   = OFF_TOPB + (size_t)B_ * C_ * PRE_NMS_ * 16;    // 81,838,080
    const size_t OFF_CI   = OFF_CS   + (size_t)B_ * C_ * POST_NMS_ * 4;    // 81,982,080

    float4*   anchors  = (float4*)(ws + OFF_ANCH);
    float4*   boxes    = (float4*)(ws + OFF_BOX);
    unsigned* keysT    = (unsigned*)(ws + OFF_KEYS);
    float*    topv     = (float*)(ws + OFF_TOPV);
    float4*   topboxes = (float4*)(ws + OFF_TOPB);
    float*    cand_s   = (float*)(ws + OFF_CS);
    int*      cand_i   = (int*)(ws + OFF_CI);

    anchors_kernel<<<(A_ + 255) / 256, 256, 0, stream>>>(anchors);
    decode_kernel<<<(B_ * A_ + 255) / 256, 256, 0, stream>>>(
        (const float4*)deltas, anchors, boxes);
    transpose_key_kernel<<<dim3((A_ + 31) / 32, (C_ + 31) / 32, B_),
                           dim3(32, 8), 0, stream>>>(labels, keysT);
    topk_kernel<<<B_ * C_, TOPK_THREADS, 0, stream>>>(keysT, boxes, topv, topboxes);
    nms_kernel<<<B_ * C_, NMS_THREADS, 0, stream>>>(topboxes, topv, cand_s, cand_i);
    final_topk_kernel<<<B_, FIN_THREADS, 0, stream>>>(cand_s, cand_i, topboxes,
                                                      (float*)d_out);
}